// SymmetricContraction_56358560858599
// MI455X (gfx1250) — compile-verified
//
#include <hip/hip_runtime.h>

typedef __attribute__((ext_vector_type(2))) float v2f;
typedef __attribute__((ext_vector_type(8))) float v8f;

#define NB 128   // atoms
#define NC 128   // channels
#define NE 10    // elements
#define NL 16    // num_ell

// ---------------------------------------------------------------------------
// wy[k,c] = sum_e w[e,k,c] * y[b,e], stored in LDS as [c][Kpad], zero-padded.
// ---------------------------------------------------------------------------
__device__ __forceinline__ void compute_wy(float* dst, int Kpad, int K,
                                           const float* __restrict__ w,
                                           const float* ys, int tid, int nthr) {
  for (int idx = tid; idx < NC * Kpad; idx += nthr) {
    int c = idx / Kpad;
    int k = idx - c * Kpad;
    float v = 0.f;
    if (k < K) {
#pragma unroll
      for (int e = 0; e < NE; ++e) v += w[(e * K + k) * NC + c] * ys[e];
    }
    dst[idx] = v;  // [c][Kpad]
  }
}

// ---------------------------------------------------------------------------
// One fused cascade (0e, or one 1o equivariant component) for a 16-channel
// tile; one wave per call. Order-3 T-tiles AND order-2 c2-tiles both via
// v_wmma_f32_16x16x4_f32, consumed straight from accumulator registers
// (lane = channel / row-half split at lane 16). The cascade is linear in the
// tile values, so each half-wave accumulates its own partial and a single
// shfl_xor(16) at the end recovers the full result (c1 pre-scaled by 0.5,
// c2 counted once per half by construction).
// ---------------------------------------------------------------------------
template <int K3, int K3PAD, int K2, int K2PAD>
__device__ __forceinline__ float run_cascade(const float* __restrict__ U3p,
                                             const float* __restrict__ U2p,
                                             const float* __restrict__ U1p,
                                             const float* wy3s,  // LDS [c][K3PAD]
                                             const float* wy2s,  // LDS [c][K2PAD]
                                             const float* wy1s,  // LDS [c]
                                             const float* xs,    // LDS [c][NL]
                                             int c, int cl16, int hi) {
  // x[c, hi*8 + j] : this lane's i-half, matches C/D layout (VGPR j -> M=j+8*hi)
  const float* xrow = xs + c * NL;
  float xreg[8];
#pragma unroll
  for (int j = 0; j < 8; ++j) xreg[j] = xrow[hi * 8 + j];

  const float wy1h = 0.5f * wy1s[c];  // c1 term is duplicated across halves

  float accTh = 0.f;
#pragma unroll 1
  for (int W = 0; W < NL; ++W) {
    // ---- order-2 tile: c2[X,c] = sum_k U2[W,X,k] * wy2[k,c]  (independent chain)
    v8f acc2 = {};
    {
      const float* A2row = U2p + (W * NL + cl16) * K2;  // A2[M=X=cl16, k]
#pragma unroll
      for (int kc = 0; kc < K2PAD / 4; ++kc) {
        int k0 = kc * 4 + hi * 2;
        int ka = k0, kb = k0 + 1;
        if (kc * 4 + 3 >= K2) {  // compile-time guarded clamp; pad cols of wy2 are 0
          ka = ka < K2 ? ka : K2 - 1;
          kb = kb < K2 ? kb : K2 - 1;
        }
        v2f a2;
        a2.x = A2row[ka];
        a2.y = A2row[kb];
        v2f b2 = *(const v2f*)(wy2s + c * K2PAD + k0);
        acc2 = __builtin_amdgcn_wmma_f32_16x16x4_f32(false, a2, false, b2,
                                                     (short)0, acc2, false, false);
      }
    }

    // ---- order-3: 16 T-tiles, dual independent WMMA chains (unroll 2)
    float accw = 0.f;
#pragma unroll 2
    for (int X = 0; X < NL; ++X) {
      v8f acc = {};
      const float* Arow = U3p + ((W * NL + X) * NL + cl16) * K3;  // A[M=cl16, k]
      __builtin_prefetch(Arow + NL * K3, 0, 3);  // next X row -> near-scope prefetch
      const float* Brow = wy3s + c * K3PAD;      // B[:, N=c] (hoisted by LICM)
#pragma unroll
      for (int kc = 0; kc < K3PAD / 4; ++kc) {
        int k0 = kc * 4 + hi * 2;
        int ka = k0, kb = k0 + 1;
        if (kc * 4 + 3 >= K3) {  // pad rows of wy3 are 0
          ka = ka < K3 ? ka : K3 - 1;
          kb = kb < K3 ? kb : K3 - 1;
        }
        v2f a;
        a.x = Arow[ka];
        a.y = Arow[kb];
        v2f b = *(const v2f*)(Brow + k0);
        acc = __builtin_amdgcn_wmma_f32_16x16x4_f32(false, a, false, b,
                                                    (short)0, acc, false, false);
      }
      // this half's partial of tval[c] = sum_i T[i,c] * x[c,i]
      float tp = acc[0] * xreg[0];
#pragma unroll
      for (int j = 1; j < 8; ++j) tp = fmaf(acc[j], xreg[j], tp);
      accw = fmaf(xrow[X], tp, accw);  // sum_X x[c,X] * tp_half(X)
    }
    // fold in this half's 8 X-rows of the c2 tile: sum_X x[c,X] * c2[X,c]
#pragma unroll
    for (int j = 0; j < 8; ++j) accw = fmaf(acc2[j], xreg[j], accw);

    // order-1 (K1==1, half-scaled) and W-contraction
    float o1 = fmaf(U1p[W], wy1h, accw);
    accTh = fmaf(xrow[W], o1, accTh);
  }
  // single cross-half combine for the whole cascade
  return accTh + __shfl_xor(accTh, 16, 32);
}

// ---------------------------------------------------------------------------
// Block = (atom b, half pp). 16 waves; wave wv owns pair p = wv + 16*pp:
//   cascade = p>>3  (0 = 0e, 1..3 = 1o eq components), ctile = p&7.
// ---------------------------------------------------------------------------
__global__ __launch_bounds__(512) void symcon_kernel(
    const float* __restrict__ x, const float* __restrict__ y,
    const float* __restrict__ U1_0e, const float* __restrict__ U2_0e,
    const float* __restrict__ U3_0e, const float* __restrict__ U1_1o,
    const float* __restrict__ U2_1o, const float* __restrict__ U3_1o,
    const float* __restrict__ w1_0e, const float* __restrict__ w2_0e,
    const float* __restrict__ w3_0e, const float* __restrict__ w1_1o,
    const float* __restrict__ w2_1o, const float* __restrict__ w3_1o,
    float* __restrict__ out) {
  __shared__ float xs[NC * NL];
  __shared__ float ys[NE];
  __shared__ float wy3_0e[NC * 24];
  __shared__ float wy2_0e[NC * 4];
  __shared__ float wy1_0e[NC];
  __shared__ float wy3_1o[NC * 36];
  __shared__ float wy2_1o[NC * 8];
  __shared__ float wy1_1o[NC];

  const int tid = threadIdx.x;
  const int b = blockIdx.x;
  const int pp = blockIdx.y;

  if (tid < NE) ys[tid] = y[b * NE + tid];
  for (int i = tid; i < NC * NL; i += 512) xs[i] = x[b * NC * NL + i];
  __syncthreads();

  compute_wy(wy3_0e, 24, 23, w3_0e, ys, tid, 512);
  compute_wy(wy2_0e, 4, 4, w2_0e, ys, tid, 512);
  compute_wy(wy1_0e, 1, 1, w1_0e, ys, tid, 512);
  compute_wy(wy3_1o, 36, 33, w3_1o, ys, tid, 512);
  compute_wy(wy2_1o, 8, 6, w2_1o, ys, tid, 512);
  compute_wy(wy1_1o, 1, 1, w1_1o, ys, tid, 512);
  __syncthreads();

  const int wv = tid >> 5;
  const int lane = tid & 31;
  const int cl16 = lane & 15;
  const int hi = lane >> 4;

  const int p = wv + 16 * pp;  // 0..31
  const int casc = p >> 3;     // 0..3
  const int ct = p & 7;
  const int c = ct * 16 + cl16;

  if (casc == 0) {
    float r = run_cascade<23, 24, 4, 4>(U3_0e, U2_0e, U1_0e, wy3_0e, wy2_0e,
                                        wy1_0e, xs, c, cl16, hi);
    if (lane < 16) out[b * 512 + c] = r;
  } else {
    const int eq = casc - 1;
    float r = run_cascade<33, 36, 6, 8>(U3_1o + eq * (NL * NL * NL * 33),
                                        U2_1o + eq * (NL * NL * 6),
                                        U1_1o + eq * NL, wy3_1o, wy2_1o,
                                        wy1_1o, xs, c, cl16, hi);
    if (lane < 16) out[b * 512 + NC + c * 3 + eq] = r;
  }
}

extern "C" void kernel_launch(void* const* d_in, const int* in_sizes, int n_in,
                              void* d_out, int out_size, void* d_ws,
                              size_t ws_size, hipStream_t stream) {
  (void)in_sizes; (void)n_in; (void)d_ws; (void)ws_size; (void)out_size;
  const float* x     = (const float*)d_in[0];
  const float* y     = (const float*)d_in[1];
  const float* U1_0e = (const float*)d_in[2];
  const float* U2_0e = (const float*)d_in[3];
  const float* U3_0e = (const float*)d_in[4];
  const float* U1_1o = (const float*)d_in[5];
  const float* U2_1o = (const float*)d_in[6];
  const float* U3_1o = (const float*)d_in[7];
  const float* w1_0e = (const float*)d_in[8];
  const float* w2_0e = (const float*)d_in[9];
  const float* w3_0e = (const float*)d_in[10];
  const float* w1_1o = (const float*)d_in[11];
  const float* w2_1o = (const float*)d_in[12];
  const float* w3_1o = (const float*)d_in[13];
  float* out = (float*)d_out;

  dim3 grid(NB, 2);
  symcon_kernel<<<grid, 512, 0, stream>>>(x, y, U1_0e, U2_0e, U3_0e, U1_1o,
                                          U2_1o, U3_1o, w1_0e, w2_0e, w3_0e,
                                          w1_1o, w2_1o, w3_1o, out);
}